// VoxelisedLinearNetwork_47107201303340
// MI455X (gfx1250) — compile-verified
//
#include <hip/hip_runtime.h>

// CDNA5 / gfx1250: wave32, WMMA 16x16x4 F32 (full-precision matrix pipe).
typedef float v2f __attribute__((ext_vector_type(2)));
typedef float v8f __attribute__((ext_vector_type(8)));

#define IN_CH            63      // 3 + 2*10*3
#define POINTS_PER_WAVE  16
#define WAVES_PER_BLOCK  8
#define POINTS_PER_BLOCK (POINTS_PER_WAVE * WAVES_PER_BLOCK)   // 128
#define BLOCK_THREADS    (WAVES_PER_BLOCK * 32)                // 256

// enc[k] for k in [0,64): k<3 -> x_k ; k=3+f*6+trig*3+d -> sin/cos(x_d * 2^f) ; k=63 -> 0 pad
__device__ __forceinline__ float enc_val(float x0, float x1, float x2, int k) {
  if (k >= 63) return 0.0f;
  if (k < 3)  return (k == 0) ? x0 : ((k == 1) ? x1 : x2);
  int t    = k - 3;
  int f    = t / 6;
  int rem  = t - 6 * f;
  int isco = (rem >= 3) ? 1 : 0;
  int d    = isco ? (rem - 3) : rem;
  float xd = (d == 0) ? x0 : ((d == 1) ? x1 : x2);
  float arg = ldexpf(xd, f);
  return isco ? __cosf(arg) : __sinf(arg);
}

__global__ __launch_bounds__(BLOCK_THREADS) void voxlin_wmma_kernel(
    const float* __restrict__ X,          // (N,3)
    const float* __restrict__ W,          // (1536,63) row-major
    const int*   __restrict__ row_ids,    // (N,)
    const int*   __restrict__ voxel_ids,  // (N,)
    float*       __restrict__ out,        // (N,3)
    int n_points)
{
  // Diagonal staging: [wave][channel][reg*32 + lane]  -> conflict-free (stride-1 across lanes)
  __shared__ float diag[WAVES_PER_BLOCK][3][32 * 8];

  const int tid  = threadIdx.x;
  const int wave = tid >> 5;
  const int l    = tid & 31;
  const int j    = l & 15;   // matrix row/column index = point-in-tile
  const int h    = l >> 4;   // K-striping half (ISA A/B layout: lanes 16-31 carry K+2)

  const int tile = blockIdx.x * WAVES_PER_BLOCK + wave;
  const int p    = tile * POINTS_PER_WAVE + j;
  const int pc   = (p < n_points) ? p : (n_points - 1);  // clamp loads, keep EXEC uniform

  const float x0  = X[3 * (size_t)pc + 0];
  const float x1  = X[3 * (size_t)pc + 1];
  const float x2  = X[3 * (size_t)pc + 2];
  const int   vox = voxel_ids[pc];

  const float* __restrict__ wrow0 = W + (size_t)(3 * vox) * IN_CH;
  __builtin_prefetch(wrow0,             0, 3);   // global_prefetch_b8: pull gathered
  __builtin_prefetch(wrow0 +     IN_CH, 0, 3);   // W rows toward this WGP early
  __builtin_prefetch(wrow0 + 2 * IN_CH, 0, 3);

  // Precompute this lane's A-fragment values: per K-step s, lane holds
  // enc[4s + 2h] and enc[4s + 2h + 1]  (VGPR0 / VGPR1 of the 16x4 f32 A tile).
  float encv[32];
#pragma unroll
  for (int s = 0; s < 16; ++s) {
    const int k0 = 4 * s + 2 * h;
    encv[2 * s]     = enc_val(x0, x1, x2, k0);
    encv[2 * s + 1] = enc_val(x0, x1, x2, k0 + 1);
  }

  // One WMMA chain per RGB channel:  D[i][n] = sum_k enc_i[k] * W[3*v_n + c][k]
  // We need only the diagonal i == n.
#pragma unroll
  for (int c = 0; c < 3; ++c) {
    const float* __restrict__ wr = wrow0 + c * IN_CH;
    v8f acc = {};
#pragma unroll
    for (int s = 0; s < 16; ++s) {
      const int k0 = 4 * s + 2 * h;
      v2f a, b;
      a.x = encv[2 * s];
      a.y = encv[2 * s + 1];
      b.x = wr[k0];                                // k0 <= 62, always in-row
      b.y = (k0 < 62) ? wr[k0 + 1] : 0.0f;         // zero-pad K=63
      acc = __builtin_amdgcn_wmma_f32_16x16x4_f32(
          /*neg_a=*/false, a, /*neg_b=*/false, b,
          /*c_mod=*/(short)0, acc, /*reuse_a=*/false, /*reuse_b=*/false);
    }
#pragma unroll
    for (int r = 0; r < 8; ++r)
      diag[wave][c][r * 32 + l] = acc[r];
  }

  __syncthreads();

  // D 16x16 f32 layout: lanes 0-15: M=reg, N=lane ; lanes 16-31: M=reg+8, N=lane-16.
  // Diagonal (M=j,N=j):  j<8 -> (reg j, lane j) ; j>=8 -> (reg j-8, lane j+16).
  if (l < 16 && p < n_points) {
    const int srcLane = (j < 8) ? j : (j + 16);
    const int srcReg  = (j < 8) ? j : (j - 8);
    const size_t row  = (size_t)row_ids[p];
#pragma unroll
    for (int c = 0; c < 3; ++c)
      out[row * 3 + c] = diag[wave][c][srcReg * 32 + srcLane];
  }
}

extern "C" void kernel_launch(void* const* d_in, const int* in_sizes, int n_in,
                              void* d_out, int out_size, void* d_ws, size_t ws_size,
                              hipStream_t stream) {
  const float* X         = (const float*)d_in[0];
  const float* W         = (const float*)d_in[1];
  const int*   row_ids   = (const int*)d_in[2];
  const int*   voxel_ids = (const int*)d_in[3];
  float*       out       = (float*)d_out;

  const int n_points = in_sizes[0] / 3;

  // Reference builds rgb = zeros(...).at[row_ids].set(vals): zero-init output
  // (capture-legal async memset), then scatter our computed rows.
  hipMemsetAsync(d_out, 0, (size_t)out_size * sizeof(float), stream);

  const int blocks = (n_points + POINTS_PER_BLOCK - 1) / POINTS_PER_BLOCK;
  voxlin_wmma_kernel<<<blocks, BLOCK_THREADS, 0, stream>>>(
      X, W, row_ids, voxel_ids, out, n_points);
}